// YOLOv4_36215164240476
// MI455X (gfx1250) — compile-verified
//
#include <hip/hip_runtime.h>
#include <cstdint>
#include <cstddef>

// ---------------- constants ----------------
#define NLVL   3
#define NCLS   80
#define TOPK   1000
#define NBINS  4096
#define CAP_A  1280
#define CAP_B  8192
#define K3     3000      // 3 levels x TOPK
#define PADK   4096      // pow2 pad for global sort
#define NW     96        // 3000 bits -> 94 words, padded to 96
#define CONF   0.01f
#define NMS_T  0.5f

#define OUT_BOX 0
#define OUT_SC  12000
#define OUT_LB  15000
#define OUT_KP  18000

typedef __attribute__((ext_vector_type(2))) float v2f;
typedef __attribute__((ext_vector_type(8))) float v8f;

// ---------------- workspace ----------------
struct WS {
  int   hist[NLVL][NBINS];   // MUST stay first (init kernel zeroes hist+cntA+cntB)
  int   cntA[NLVL];
  int   cntB[NLVL];
  int   Bq[NLVL];
  int   nAb[NLVL];
  float off_scale;
  int   pad_;
  int   aIdx[NLVL][CAP_A];
  float aSc [NLVL][CAP_A];   // holds score^2 (monotone in score)
  int   bIdx[NLVL][CAP_B];
  float bSc [NLVL][CAP_B];   // holds score^2
  int   finIdx[NLVL][TOPK];
  float finSc [NLVL][TOPK];  // holds score^2; sqrt applied once at decode
  float cbox[K3 * 4];
  float cscore[K3];          // real score (sqrt'd)
  int   clabel[K3];
  int   perm[K3];
  float obox[K3 * 4];        // label-offset boxes (sorted order)
  float oarea[K3];
  float sscore[K3];
  unsigned keepWords[NW];
  unsigned mask[K3 * NW];    // suppression bitmask, row-major
};

struct KParams {
  const float* obj[NLVL];
  const float* cls[NLVL];
  const float* reg[NLVL];
  const float* anch;
  WS* ws;
  float* out;
};

// ---------------- helpers ----------------
// fast sigmoid: v_exp + v_add + v_rcp (raw rcp, ~1 ulp) -- used identically in
// both score passes so binning is bit-consistent between them.
__device__ inline float sigm(float x) {
  return __builtin_amdgcn_rcpf(1.0f + __expf(-x));
}
// accurate sigmoid for the 3000-element box decode
__device__ inline float sigm_acc(float x) { return 1.0f / (1.0f + expf(-x)); }

// D = diag(sig(obj[16])) x sig(cls tile 16x16), via 4 chained fp32 WMMAs.
// fp32 A 16x4 layout (ISA 7.12.2): lane m=lane&15; VGPR0 holds K=2*hi, VGPR1 K=2*hi+1.
// fp32 B 4x16 mirror layout: VGPR0 row K=2*hi, VGPR1 row K=2*hi+1, n=lane&15.
// D 16x16: acc[r] = element (M = r + 8*hi, N = lane&15).
__device__ inline v8f score2_tile(const float* __restrict__ cls, int m0, int c0, float s_m) {
  const int lane = threadIdx.x & 31;
  const int hi   = lane >> 4;
  const int n    = lane & 15;
  const int m    = lane & 15;
  v8f acc = {0.f, 0.f, 0.f, 0.f, 0.f, 0.f, 0.f, 0.f};
#pragma unroll
  for (int k = 0; k < 4; ++k) {
    const int r0 = 4 * k + 2 * hi;
    v2f a, b;
    a.x = (m == r0    ) ? s_m : 0.f;
    a.y = (m == r0 + 1) ? s_m : 0.f;
    b.x = sigm(cls[(m0 + r0    ) * NCLS + c0 + n]);
    b.y = sigm(cls[(m0 + r0 + 1) * NCLS + c0 + n]);
    acc = __builtin_amdgcn_wmma_f32_16x16x4_f32(false, a, false, b, (short)0, acc, false, false);
  }
  return acc; // acc[r] = sig(obj[m0+r+8*hi]) * sig(cls[...][c0+n])  (= score^2)
}

// ---------------- kernels ----------------
__global__ void yolo_init_kernel(WS* ws) {
  int i = blockIdx.x * 256 + threadIdx.x;
  const int N = NLVL * NBINS + 2 * NLVL;  // hist + cntA + cntB (contiguous)
  if (i < N) ((int*)ws)[i] = 0;
}

// PASS 0: per-level 4096-bin histogram over score^2 (monotone in score).
// PASS 1: compaction vs Bq (identical score^2 computation -> identical bins).
template <int PASS>
__global__ __launch_bounds__(256) void yolo_score_pass(KParams p) {
  __shared__ int lh[NBINS];
  if (PASS == 0) {
    for (int i = threadIdx.x; i < NBINS; i += 256) lh[i] = 0;
    __syncthreads();
  }
  int lvl, base;
  if (blockIdx.x < 600)      { lvl = 0; base = 0;   }
  else if (blockIdx.x < 750) { lvl = 1; base = 600; }
  else                       { lvl = 2; base = 750; }
  const int nStrips = (lvl == 0) ? 4800 : ((lvl == 1) ? 1200 : 300);
  const int wave  = threadIdx.x >> 5;
  const int strip = ((int)blockIdx.x - base) * 8 + wave;
  WS* ws = p.ws;
  if (strip < nStrips) {  // wave-uniform; EXEC stays all-1s for WMMA
    const float* obj = p.obj[lvl];
    const float* cls = p.cls[lvl];
    const int m0   = strip * 16;
    const int lane = threadIdx.x & 31;
    const int hi   = lane >> 4;
    const int n    = lane & 15;
    const float s_m = sigm(obj[m0 + (lane & 15)]);
    const int Bq = (PASS == 1) ? ws->Bq[lvl] : 0;
    for (int c0 = 0; c0 < NCLS; c0 += 16) {
      v8f acc = score2_tile(cls, m0, c0, s_m);
#pragma unroll
      for (int r = 0; r < 8; ++r) {
        float s2 = acc[r];                      // score^2 in (0,1)
        int bin = (int)(s2 * (float)NBINS);
        bin = bin < 0 ? 0 : (bin > NBINS - 1 ? NBINS - 1 : bin);
        if (PASS == 0) {
          atomicAdd(&lh[bin], 1);
        } else if (bin >= Bq) {
          int idx = (m0 + r + 8 * hi) * NCLS + (c0 + n);
          if (bin > Bq) {
            int pos = atomicAdd(&ws->cntA[lvl], 1);
            if (pos < CAP_A) { ws->aIdx[lvl][pos] = idx; ws->aSc[lvl][pos] = s2; }
          } else {
            int pos = atomicAdd(&ws->cntB[lvl], 1);
            if (pos < CAP_B) { ws->bIdx[lvl][pos] = idx; ws->bSc[lvl][pos] = s2; }
          }
        }
      }
    }
  }
  if (PASS == 0) {
    __syncthreads();
    for (int i = threadIdx.x; i < NBINS; i += 256) {
      int c = lh[i];
      if (c) atomicAdd(&p.ws->hist[lvl][i], c);
    }
  }
}

__global__ void yolo_threshold_kernel(KParams p) {
  int lvl = threadIdx.x;
  if (lvl >= NLVL) return;
  WS* ws = p.ws;
  int running = 0;
  for (int b = NBINS - 1; b >= 0; --b) {
    int c = ws->hist[lvl][b];
    if (running + c >= TOPK) { ws->Bq[lvl] = b; ws->nAb[lvl] = running; return; }
    running += c;
  }
  ws->Bq[lvl] = 0;
  ws->nAb[lvl] = running;
}

// descending bitonic sort on LDS (key,payload)
__device__ void bitonic_desc(float* k, int* v, int P, int tid, int nthr) {
  for (int size = 2; size <= P; size <<= 1) {
    for (int stride = size >> 1; stride > 0; stride >>= 1) {
      __syncthreads();
      for (int i = tid; i < P; i += nthr) {
        int partner = i ^ stride;
        if (partner > i) {
          bool dirDesc = ((i & size) == 0);
          float ki = k[i], kp = k[partner];
          bool sw = dirDesc ? (ki < kp) : (ki > kp);
          if (sw) {
            k[i] = kp; k[partner] = ki;
            int t = v[i]; v[i] = v[partner]; v[partner] = t;
          }
        }
      }
    }
  }
  __syncthreads();
}

// per level: above-list + top slice of sorted boundary bin -> exactly TOPK finalists
__global__ __launch_bounds__(1024) void yolo_finalize_kernel(KParams p) {
  extern __shared__ char smf[];
  float* sc = (float*)smf;
  int*   id = (int*)(smf + CAP_B * sizeof(float));
  const int lvl = blockIdx.x, tid = threadIdx.x;
  WS* ws = p.ws;
  int nA = ws->cntA[lvl];
  if (nA > CAP_A) nA = CAP_A;
  if (nA > TOPK)  nA = TOPK;
  for (int i = tid; i < nA; i += 1024) {
    ws->finIdx[lvl][i] = ws->aIdx[lvl][i];
    ws->finSc [lvl][i] = ws->aSc [lvl][i];
  }
  int nB = ws->cntB[lvl];
  if (nB > CAP_B) nB = CAP_B;
  for (int i = tid; i < CAP_B; i += 1024) {
    sc[i] = (i < nB) ? ws->bSc[lvl][i] : -1e30f;
    id[i] = (i < nB) ? ws->bIdx[lvl][i] : 0;
  }
  bitonic_desc(sc, id, CAP_B, tid, 1024);
  const int need = TOPK - nA;
  for (int i = tid; i < need; i += 1024) {
    float s = sc[i]; int ix = id[i];
    if (s < -1e29f) { s = 0.f; ix = 0; }  // underflow fill (never valid)
    ws->finIdx[lvl][nA + i] = ix;
    ws->finSc [lvl][nA + i] = s;
  }
}

__global__ __launch_bounds__(256) void yolo_decode_kernel(KParams p) {
  int g = blockIdx.x * 256 + threadIdx.x;
  if (g >= K3) return;
  WS* ws = p.ws;
  const int lvl = g / TOPK;
  const int j   = g - lvl * TOPK;
  const int idx = ws->finIdx[lvl][j];
  const float sc = sqrtf(ws->finSc[lvl][j]);  // one accurate sqrt per finalist
  const int anchor = idx / NCLS;
  const int label  = idx - anchor * NCLS;
  const int a    = anchor % 3;
  const int cell = anchor / 3;
  const int   Wt[NLVL] = {160, 80, 40};
  const float St[NLVL] = {8.f, 16.f, 32.f};
  const int Wl = Wt[lvl];
  const int x = cell % Wl, y = cell / Wl;
  const float* reg = p.reg[lvl] + (size_t)anchor * 4;
  float cx = (sigm_acc(reg[0]) * 3.f - 1.5f + (float)x + 0.5f) * St[lvl];
  float cy = (sigm_acc(reg[1]) * 3.f - 1.5f + (float)y + 0.5f) * St[lvl];
  float aw = p.anch[(lvl * 3 + a) * 2 + 0];
  float ah = p.anch[(lvl * 3 + a) * 2 + 1];
  float w = expf(reg[2]) * aw, h = expf(reg[3]) * ah;
  ws->cbox[g * 4 + 0] = cx - 0.5f * w;
  ws->cbox[g * 4 + 1] = cy - 0.5f * h;
  ws->cbox[g * 4 + 2] = cx + 0.5f * w;
  ws->cbox[g * 4 + 3] = cy + 0.5f * h;
  ws->cscore[g] = sc;
  ws->clabel[g] = label;
}

// global sort of 3000 candidates by (valid?score:-1), desc; also off_scale = max(boxes)+1
__global__ __launch_bounds__(1024) void yolo_sort_kernel(KParams p) {
  extern __shared__ char sms[];
  float* key = (float*)sms;
  int*   pay = (int*)(sms + PADK * sizeof(float));
  __shared__ float red[1024];
  WS* ws = p.ws;
  const int tid = threadIdx.x;
  float mx = -1e30f;
  for (int i = tid; i < K3 * 4; i += 1024) mx = fmaxf(mx, ws->cbox[i]);
  red[tid] = mx; __syncthreads();
  for (int s = 512; s > 0; s >>= 1) {
    if (tid < s) red[tid] = fmaxf(red[tid], red[tid + s]);
    __syncthreads();
  }
  if (tid == 0) ws->off_scale = red[0] + 1.0f;
  for (int i = tid; i < PADK; i += 1024) {
    if (i < K3) {
      float sc = ws->cscore[i];
      key[i] = (sc > CONF) ? sc : -1.0f;
      pay[i] = i;
    } else { key[i] = -2.0f; pay[i] = 0; }
  }
  bitonic_desc(key, pay, PADK, tid, 1024);
  for (int i = tid; i < K3; i += 1024) ws->perm[i] = pay[i];
}

// gather sorted order, apply label offset (reference's class-aware trick), emit boxes/labels,
// and build initial keep bitmask via wave ballot
__global__ __launch_bounds__(256) void yolo_prep_kernel(KParams p) {
  int j = blockIdx.x * 256 + threadIdx.x;  // 0..3071
  WS* ws = p.ws;
  int valid = 0;
  if (j < K3) {
    const int pi = ws->perm[j];
    float b0 = ws->cbox[pi * 4 + 0], b1 = ws->cbox[pi * 4 + 1];
    float b2 = ws->cbox[pi * 4 + 2], b3 = ws->cbox[pi * 4 + 3];
    const int lab = ws->clabel[pi];
    const float sc = ws->cscore[pi];
    const float off = ws->off_scale * (float)lab;
    ws->obox[j * 4 + 0] = b0 + off; ws->obox[j * 4 + 1] = b1 + off;
    ws->obox[j * 4 + 2] = b2 + off; ws->obox[j * 4 + 3] = b3 + off;
    ws->oarea[j] = (b2 - b0) * (b3 - b1);
    ws->sscore[j] = sc;
    p.out[OUT_BOX + j * 4 + 0] = b0; p.out[OUT_BOX + j * 4 + 1] = b1;
    p.out[OUT_BOX + j * 4 + 2] = b2; p.out[OUT_BOX + j * 4 + 3] = b3;
    p.out[OUT_LB + j] = (float)lab;
    valid = (sc > CONF) ? 1 : 0;
  }
  unsigned long long bal = __ballot(valid);
  if ((threadIdx.x & 31) == 0) ws->keepWords[j >> 5] = (unsigned)bal;
}

// suppression bitmask: row i, bit j set iff j>i and IoU(offset boxes) > 0.5
__global__ __launch_bounds__(128) void yolo_mask_kernel(KParams p) {
  WS* ws = p.ws;
  const int i = blockIdx.x;
  const float ax1 = ws->obox[i * 4 + 0], ay1 = ws->obox[i * 4 + 1];
  const float ax2 = ws->obox[i * 4 + 2], ay2 = ws->obox[i * 4 + 3];
  const float aa  = ws->oarea[i];
  for (int j0 = 0; j0 < NW * 32; j0 += 128) {
    const int j = j0 + threadIdx.x;
    int pred = 0;
    if (j < K3 && j > i) {
      float xx1 = fmaxf(ax1, ws->obox[j * 4 + 0]);
      float yy1 = fmaxf(ay1, ws->obox[j * 4 + 1]);
      float xx2 = fminf(ax2, ws->obox[j * 4 + 2]);
      float yy2 = fminf(ay2, ws->obox[j * 4 + 3]);
      float iw = fmaxf(xx2 - xx1, 0.f), ih = fmaxf(yy2 - yy1, 0.f);
      float inter = iw * ih;
      float uni = aa + ws->oarea[j] - inter;
      pred = (inter > NMS_T * fmaxf(uni, 1e-9f)) ? 1 : 0;  // iou > T
    }
    unsigned long long bal = __ballot(pred);
    if ((threadIdx.x & 31) == 0)
      ws->mask[i * NW + (j0 >> 5) + (threadIdx.x >> 5)] = (unsigned)bal;
  }
}

// sequential greedy resolve on a single wave32; 8-deep prefetch of mask rows.
__global__ void yolo_scan_kernel(KParams p) {
  WS* ws = p.ws;
  const int lane = threadIdx.x & 31;
  unsigned k0 = ws->keepWords[lane];
  unsigned k1 = ws->keepWords[lane + 32];
  unsigned k2 = ws->keepWords[lane + 64];
  unsigned b0[8], b1[8], b2[8];
#pragma unroll
  for (int d = 0; d < 8; ++d) {
    b0[d] = ws->mask[d * NW + lane];
    b1[d] = ws->mask[d * NW + lane + 32];
    b2[d] = ws->mask[d * NW + lane + 64];
  }
  for (int ib = 0; ib < K3; ib += 8) {   // K3 % 8 == 0
#pragma unroll
    for (int u = 0; u < 8; ++u) {
      const int i = ib + u;
      const unsigned r0 = b0[u], r1 = b1[u], r2 = b2[u];
      const int ip = i + 8;
      if (ip < K3) {
        b0[u] = ws->mask[ip * NW + lane];
        b1[u] = ws->mask[ip * NW + lane + 32];
        b2[u] = ws->mask[ip * NW + lane + 64];
      }
      const int wi = i >> 5, s = wi >> 5, owner = wi & 31;
      unsigned kw = (s == 0) ? k0 : ((s == 1) ? k1 : k2);
      kw = (unsigned)__shfl((int)kw, owner, 32);
      if ((kw >> (i & 31)) & 1u) { k0 &= ~r0; k1 &= ~r1; k2 &= ~r2; }
    }
  }
  ws->keepWords[lane]      = k0;
  ws->keepWords[lane + 32] = k1;
  ws->keepWords[lane + 64] = k2;
}

__global__ __launch_bounds__(256) void yolo_writeout_kernel(KParams p) {
  int i = blockIdx.x * 256 + threadIdx.x;
  if (i >= K3) return;
  WS* ws = p.ws;
  unsigned kb = (ws->keepWords[i >> 5] >> (i & 31)) & 1u;
  p.out[OUT_SC + i] = kb ? ws->sscore[i] : 0.f;
  p.out[OUT_KP + i] = (float)kb;
}

// ---------------- launcher ----------------
extern "C" void kernel_launch(void* const* d_in, const int* in_sizes, int n_in,
                              void* d_out, int out_size, void* d_ws, size_t ws_size,
                              hipStream_t stream) {
  (void)in_sizes; (void)n_in; (void)out_size; (void)ws_size;  // WS ~1.6MB assumed available
  KParams p;
  for (int l = 0; l < NLVL; ++l) {
    p.obj[l] = (const float*)d_in[3 * l + 0];
    p.cls[l] = (const float*)d_in[3 * l + 1];
    p.reg[l] = (const float*)d_in[3 * l + 2];
  }
  p.anch = (const float*)d_in[9];
  p.ws   = (WS*)d_ws;
  p.out  = (float*)d_out;

  const int initN = NLVL * NBINS + 2 * NLVL;
  yolo_init_kernel<<<(initN + 255) / 256, 256, 0, stream>>>(p.ws);
  yolo_score_pass<0><<<788, 256, 0, stream>>>(p);          // 600 + 150 + 38 blocks
  yolo_threshold_kernel<<<1, 32, 0, stream>>>(p);
  yolo_score_pass<1><<<788, 256, 0, stream>>>(p);
  yolo_finalize_kernel<<<NLVL, 1024, CAP_B * 8, stream>>>(p);
  yolo_decode_kernel<<<(K3 + 255) / 256, 256, 0, stream>>>(p);
  yolo_sort_kernel<<<1, 1024, PADK * 8, stream>>>(p);
  yolo_prep_kernel<<<(NW * 32) / 256, 256, 0, stream>>>(p);  // 12 blocks (3072 threads)
  yolo_mask_kernel<<<K3, 128, 0, stream>>>(p);
  yolo_scan_kernel<<<1, 32, 0, stream>>>(p);
  yolo_writeout_kernel<<<(K3 + 255) / 256, 256, 0, stream>>>(p);
}